// Rank_79826262163819
// MI455X (gfx1250) — compile-verified
//
#include <hip/hip_runtime.h>

#define NSTIM   10000
#define NDIM    128
#define NGROUP  4
#define BATCH   65536
#define NREF    8
#define BETA    10.0f

typedef __attribute__((ext_vector_type(2))) float v2f;
typedef __attribute__((ext_vector_type(8))) float v8f;
typedef __attribute__((address_space(3))) float lds_f32;

__global__ __launch_bounds__(256) void Rank_79826262163819_kernel(
    const int*           __restrict__ stim,        // BATCH x (NREF+1)
    const int*           __restrict__ membership,  // BATCH x 2
    const unsigned char* __restrict__ is_present,  // BATCH x (NREF+1)  (bool)
    const unsigned char* __restrict__ is_select,   // BATCH x 2         (bool)
    const float*         __restrict__ z,           // NSTIM x NDIM
    const float*         __restrict__ w,           // NGROUP x NDIM
    float*               __restrict__ out)         // BATCH
{
    __shared__ float wbuf[NGROUP * NDIM];   // 2KB: attention weights, block-wide
    __shared__ float zqbuf[8][2 * NDIM];    // 8KB: per-wave query rows (2 trials)
    __shared__ float dmat[8][16][16];       // 8KB: per-wave D matrix spill
    __shared__ float simbuf[8][16];         // per-wave similarities

    const int tid  = threadIdx.x;
    const int wave = tid >> 5;          // 0..7
    const int lane = tid & 31;
    const int half = lane >> 4;         // 0: K={0,1}, 1: K={2,3} of each chunk
    const int lm   = lane & 15;         // pair row M (and column N)

    // two trials per wave; pair m -> (trial_local = m>>3, ref = m&7)
    const int t0    = (blockIdx.x * 8 + wave) * 2;
    const int trial = t0 + (lm >> 3);
    const int ref   = lm & 7;

    // ---- stage w into LDS (block-wide, coalesced) ----
    for (int i = tid; i < NGROUP * NDIM; i += 256)
        wbuf[i] = w[i];

    // ---- stage the wave's two query rows into LDS via async loads ----
    {
        const int zq0 = stim[(t0 + 0) * (NREF + 1)];
        const int zq1 = stim[(t0 + 1) * (NREF + 1)];
        const float* g0 = z + (size_t)zq0 * NDIM + lane * 4;   // 16B per lane
        const float* g1 = z + (size_t)zq1 * NDIM + lane * 4;
        lds_f32* l0 = (lds_f32*)&zqbuf[wave][lane * 4];
        lds_f32* l1 = (lds_f32*)&zqbuf[wave][NDIM + lane * 4];
        asm volatile("global_load_async_to_lds_b128 %0, %1, off"
                     :: "v"(l0), "v"(g0) : "memory");
        asm volatile("global_load_async_to_lds_b128 %0, %1, off"
                     :: "v"(l1), "v"(g1) : "memory");
    }

    // z_r row for this pair (unique -> direct global gather)
    const int zr_idx = stim[trial * (NREF + 1) + 1 + ref];
    const float* __restrict__ zr = z + (size_t)zr_idx * NDIM;

    asm volatile("s_wait_asynccnt 0x0" ::: "memory");
    __syncthreads();   // wbuf + zqbuf visible

    const float* __restrict__ zq_l = &zqbuf[wave][(lm >> 3) * NDIM];
    const float* __restrict__ w_l  = &wbuf[(lm & (NGROUP - 1)) * NDIM];

    // ---- K-loop: 32 chunks of 4 dims, f32 WMMA accumulate (EXEC all-ones) ----
    v8f acc = {};
#pragma unroll 4
    for (int c = 0; c < NDIM / 4; ++c) {
        const int kb = c * 4 + half * 2;
        const float q0 = zq_l[kb], q1 = zq_l[kb + 1];   // ds_load_b64 (broadcast)
        const float r0 = zr[kb],   r1 = zr[kb + 1];     // global_load_b64 (gather)
        const float d0 = q0 - r0;
        const float d1 = q1 - r1;
        v2f a; a.x = d0 * d0;    a.y = d1 * d1;         // |diff|^RHO, RHO=2
        v2f b; b.x = w_l[kb];    b.y = w_l[kb + 1];     // group weight columns
        acc = __builtin_amdgcn_wmma_f32_16x16x4_f32(
            /*neg_a=*/false, a, /*neg_b=*/false, b,
            /*c_mod=*/(short)0, acc, /*reuse_a=*/false, /*reuse_b=*/false);
    }

    // spill D: VGPR r holds row M = r + 8*half, column N = lm
#pragma unroll
    for (int r = 0; r < 8; ++r)
        dmat[wave][r + 8 * half][lm] = acc[r];
    __syncthreads();

    // lanes 0..15: pair m = lm reads d^2 at its trial's group column
    if (half == 0) {
        const int g   = membership[trial * 2 + 0] & (NGROUP - 1);
        const float d2 = dmat[wave][lm][g];
        const float d  = sqrtf(d2);                     // ^(1/RHO)
        float sim = expf(-BETA * d);                    // TAU=1, GAMMA=0
        sim *= (float)is_present[trial * (NREF + 1) + 1 + ref];
        simbuf[wave][lm] = sim;
    }
    __syncthreads();

    // one lane per trial: suffix-sum denominators, 2-term selection product
    if (half == 0 && (lm & 7) == 0) {
        const int tl = lm >> 3;                          // 0 or 1
        const int tr = t0 + tl;
        const float* __restrict__ s = &simbuf[wave][tl * 8];
        float den[NREF];
        float suffix = 0.0f;
        for (int r = NREF - 1; r >= 0; --r) { suffix += s[r]; den[r] = suffix; }
        float prob = 1.0f;
#pragma unroll
        for (int i = 0; i < 2; ++i) {
            const float dn  = den[i];
            const float p   = (dn == 0.0f) ? 0.0f : s[i] / dn;
            const float sel = (float)is_select[tr * 2 + i];
            prob *= p * sel + (1.0f - sel);
        }
        out[tr] = prob;
    }
}

extern "C" void kernel_launch(void* const* d_in, const int* in_sizes, int n_in,
                              void* d_out, int out_size, void* d_ws, size_t ws_size,
                              hipStream_t stream) {
    (void)in_sizes; (void)n_in; (void)out_size; (void)d_ws; (void)ws_size;
    const int*           stim       = (const int*)d_in[0];
    const int*           membership = (const int*)d_in[1];
    const unsigned char* is_present = (const unsigned char*)d_in[2];
    const unsigned char* is_select  = (const unsigned char*)d_in[3];
    const float*         z          = (const float*)d_in[4];
    const float*         w          = (const float*)d_in[5];
    float*               out        = (float*)d_out;

    // 8 waves/block, 2 trials/wave -> 16 trials/block; 65536/16 = 4096 blocks
    Rank_79826262163819_kernel<<<BATCH / 16, 256, 0, stream>>>(
        stim, membership, is_present, is_select, z, w, out);
}